// Gaussian_83270825934914
// MI455X (gfx1250) — compile-verified
//
#include <hip/hip_runtime.h>
#include <hip/hip_bf16.h>
#include <math.h>

#define DIM   512
#define BATCH 256
#define PW    32                  // panel width (K per trailing update)
#define NPB   (DIM / PW)          // 16 panels
#define LOG2PI 1.8378770664093453f

typedef __attribute__((ext_vector_type(2))) float v2f;
typedef __attribute__((ext_vector_type(8))) float v8f;

// ---------------------------------------------------------------------------
// Kernel 1: A[i][j] = lower_tri (j<i), exp(log_diag) (j==i), 0 (j>i)
// ---------------------------------------------------------------------------
__global__ void build_A_kernel(const float* __restrict__ logd,
                               const float* __restrict__ lt,
                               float* __restrict__ A) {
  const int i = blockIdx.x;
  for (int j = threadIdx.x; j < DIM; j += blockDim.x) {
    float v;
    if (j < i)       v = lt[(size_t)i * (i - 1) / 2 + j];
    else if (j == i) v = expf(logd[i]);
    else             v = 0.0f;
    A[(size_t)i * DIM + j] = v;
  }
}

// ---------------------------------------------------------------------------
// Kernel 2: cov = A * A^T (k runs only to min(i,j) since A is lower-tri)
// ---------------------------------------------------------------------------
__global__ void build_cov_kernel(const float* __restrict__ A,
                                 float* __restrict__ cov) {
  __shared__ float arow[DIM];
  const int i = blockIdx.x;
  for (int j = threadIdx.x; j < DIM; j += blockDim.x)
    arow[j] = A[(size_t)i * DIM + j];
  __syncthreads();
  for (int j = threadIdx.x; j < DIM; j += blockDim.x) {
    const int kmax = (j < i) ? j : i;
    const float* Aj = &A[(size_t)j * DIM];
    float s = 0.0f;
    for (int k = 0; k <= kmax; ++k) s += arow[k] * Aj[k];
    cov[(size_t)i * DIM + j] = s;
  }
}

// ---------------------------------------------------------------------------
// Kernel 3: per-sample masked Cholesky (blocked, panel width 32) + solve + NLL
// One workgroup (256 threads = 8 wave32) per sample; slab = 512x512 f32 in ws.
// Trailing SYRK: 16x16 C tiles, 8 chained V_WMMA_F32_16X16X4_F32 (K=32) each.
// ---------------------------------------------------------------------------
__global__ __launch_bounds__(256)
void chol_nll_kernel(const float* __restrict__ cov,
                     const float* __restrict__ x,
                     const float* __restrict__ mu,
                     const unsigned char* __restrict__ mask,
                     float* __restrict__ out,
                     float* __restrict__ slabs,
                     int sample_base) {
  __shared__ float lds_kk[PW * PW];        // diagonal block       (4 KB)
  __shared__ float lds_panel[DIM * PW];    // current column panel (64 KB)
  __shared__ float lds_m[DIM];             // mask as float
  __shared__ float lds_rz[DIM];            // residual -> z
  __shared__ float red[256];

  const int tid = threadIdx.x;
  const int s   = sample_base + blockIdx.x;
  float* slab = slabs + (size_t)blockIdx.x * DIM * DIM;

  // ---- mask row to LDS ----------------------------------------------------
  const unsigned char* mrow = mask + (size_t)s * DIM;
  for (int i = tid; i < DIM; i += 256) lds_m[i] = mrow[i] ? 1.0f : 0.0f;
  __syncthreads();

  // ---- build masked covariance into the slab ------------------------------
  // diag: m_i*cov_ii + (1-m_i) ; off-diag: m_i*m_j*cov_ij
  for (int idx = tid; idx < DIM * DIM; idx += 256) {
    const int i = idx >> 9, j = idx & (DIM - 1);
    const float mi = lds_m[i], mj = lds_m[j];
    const float cij = cov[idx];
    slab[idx] = (i == j) ? (mi * cij + (1.0f - mi)) : (mi * mj * cij);
  }
  __threadfence_block();
  __syncthreads();

  const int wave = tid >> 5;
  const int lane = tid & 31;
  const int half = lane >> 4;    // 0: lanes 0-15, 1: lanes 16-31
  const int ln   = lane & 15;

  // ---- blocked right-looking Cholesky (panel width PW=32) -----------------
  for (int kb = 0; kb < NPB; ++kb) {
    const int k0 = kb * PW;

    // 1) load diagonal 32x32 tile into LDS
    for (int idx = tid; idx < PW * PW; idx += 256)
      lds_kk[idx] = slab[(size_t)(k0 + (idx >> 5)) * DIM + k0 + (idx & 31)];
    __syncthreads();

    // 2) factor diagonal tile (tiny: thread 0)
    if (tid == 0) {
      for (int c = 0; c < PW; ++c) {
        float d = lds_kk[c * PW + c];
        for (int t = 0; t < c; ++t) d -= lds_kk[c * PW + t] * lds_kk[c * PW + t];
        d = sqrtf(d);
        lds_kk[c * PW + c] = d;
        const float inv = 1.0f / d;
        for (int r = c + 1; r < PW; ++r) {
          float v = lds_kk[r * PW + c];
          for (int t = 0; t < c; ++t) v -= lds_kk[r * PW + t] * lds_kk[c * PW + t];
          lds_kk[r * PW + c] = v * inv;
        }
      }
    }
    __syncthreads();

    // write factored lower part back to slab (diag needed for logdet/solve)
    for (int idx = tid; idx < PW * PW; idx += 256) {
      const int r = idx >> 5, c = idx & 31;
      if (c <= r) slab[(size_t)(k0 + r) * DIM + k0 + c] = lds_kk[idx];
    }

    // 3) panel solve: rows below, L[i,:] = A[i,:] * inv(Lkk)^T ; cache in LDS
    const int rs = k0 + PW;
    for (int i = rs + tid; i < DIM; i += 256) {
      float v[PW];
      float* row = &slab[(size_t)i * DIM + k0];
      #pragma unroll
      for (int c = 0; c < PW; ++c) v[c] = row[c];
      #pragma unroll
      for (int c = 0; c < PW; ++c) {
        float sum = v[c];
        for (int t = 0; t < c; ++t) sum -= v[t] * lds_kk[c * PW + t];
        v[c] = sum / lds_kk[c * PW + c];
      }
      float* prow = &lds_panel[i * PW];
      #pragma unroll
      for (int c = 0; c < PW; ++c) { row[c] = v[c]; prow[c] = v[c]; }
    }
    __threadfence_block();
    __syncthreads();

    // 4) trailing SYRK update: C[bi,bj] -= Lp[bi] * Lp[bj]^T  via f32 WMMA
    const int r0  = k0 + PW;               // first trailing row
    const int b0  = r0 >> 4;               // first trailing 16-block
    const int T16 = (DIM - r0) >> 4;       // # of 16-row blocks remaining
    if (T16 > 0) {
      const int nt = T16 * (T16 + 1) / 2;  // lower-tri 16x16 tiles
      for (int t = wave; t < nt; t += 8) { // wave-uniform loop
        int rr = 0, acc = 0;
        while (acc + rr + 1 <= t) { acc += rr + 1; ++rr; }   // uniform decode
        const int bi = __builtin_amdgcn_readfirstlane(b0 + rr);
        const int bj = __builtin_amdgcn_readfirstlane(b0 + (t - acc));

        float* ctile = &slab[(size_t)(bi * 16) * DIM + bj * 16];
        // C/D layout: VGPR v -> M = v + 8*half, N = ln
        v8f c;
        #pragma unroll
        for (int vv = 0; vv < 8; ++vv)
          c[vv] = ctile[(size_t)(vv + 8 * half) * DIM + ln];

        const float* pA = &lds_panel[(bi * 16 + ln) * PW]; // A row M = ln
        const float* pB = &lds_panel[(bj * 16 + ln) * PW]; // B^T col N = ln
        #pragma unroll
        for (int kk = 0; kk < 8; ++kk) {
          // A 16x4 layout: K = 2*half + {0,1} within each 4-slice
          v2f a, b;
          a[0] = -pA[kk * 4 + 2 * half + 0];   // negate A => C - A*B^T
          a[1] = -pA[kk * 4 + 2 * half + 1];
          b[0] =  pB[kk * 4 + 2 * half + 0];
          b[1] =  pB[kk * 4 + 2 * half + 1];
          c = __builtin_amdgcn_wmma_f32_16x16x4_f32(
                  /*neg_a=*/false, a, /*neg_b=*/false, b,
                  /*c_mod=*/(short)0, c, /*reuse_a=*/false, /*reuse_b=*/false);
        }
        #pragma unroll
        for (int vv = 0; vv < 8; ++vv)
          ctile[(size_t)(vv + 8 * half) * DIM + ln] = c[vv];
      }
    }
    __threadfence_block();
    __syncthreads();
  }

  // ---- masked residual ----------------------------------------------------
  for (int i = tid; i < DIM; i += 256)
    lds_rz[i] = lds_m[i] * (x[(size_t)s * DIM + i] - mu[i]);
  __syncthreads();

  // ---- blocked forward substitution  L z = r ------------------------------
  for (int kb = 0; kb < NPB; ++kb) {
    const int k0 = kb * PW;
    if (tid == 0) {
      for (int c = 0; c < PW; ++c) {
        const int j = k0 + c;
        const float* Lr = &slab[(size_t)j * DIM + k0];
        float v = lds_rz[j];
        for (int t = 0; t < c; ++t) v -= Lr[t] * lds_rz[k0 + t];
        lds_rz[j] = v / slab[(size_t)j * DIM + j];
      }
    }
    __syncthreads();
    for (int i = k0 + PW + tid; i < DIM; i += 256) {
      const float* Lr = &slab[(size_t)i * DIM + k0];
      float v = lds_rz[i];
      #pragma unroll
      for (int c = 0; c < PW; ++c) v -= Lr[c] * lds_rz[k0 + c];
      lds_rz[i] = v;
    }
    __syncthreads();
  }

  // ---- reduce: 0.5 * (sum z^2 + 2 sum log Lii + d_obs*log(2pi)) -----------
  float part = 0.0f;
  for (int i = tid; i < DIM; i += 256) {
    const float z = lds_rz[i];
    const float Lii = slab[(size_t)i * DIM + i];   // unobserved: Lii==1 -> log 0
    part += z * z + 2.0f * logf(Lii) + lds_m[i] * LOG2PI;
  }
  red[tid] = part;
  __syncthreads();
  for (int st = 128; st > 0; st >>= 1) {
    if (tid < st) red[tid] += red[tid + st];
    __syncthreads();
  }
  if (tid == 0) out[s] = 0.5f * red[0];
}

// ---------------------------------------------------------------------------
// Host launcher
// ---------------------------------------------------------------------------
extern "C" void kernel_launch(void* const* d_in, const int* in_sizes, int n_in,
                              void* d_out, int out_size, void* d_ws, size_t ws_size,
                              hipStream_t stream) {
  const float*         x    = (const float*)d_in[0];          // [256,512]
  const float*         mu   = (const float*)d_in[1];          // [512]
  const float*         logd = (const float*)d_in[2];          // [512]
  const float*         lt   = (const float*)d_in[3];          // [130816]
  const unsigned char* mask = (const unsigned char*)d_in[4];  // [256,512] bool
  float* out = (float*)d_out;                                 // [256]

  const size_t MAT = (size_t)DIM * DIM;       // elements per 512x512 matrix
  float* ws    = (float*)d_ws;
  float* dA    = ws;                           // 1 MB
  float* dcov  = ws + MAT;                     // 1 MB
  float* slabs = ws + 2 * MAT;                 // P x 1 MB sample slabs

  long long availElems = (long long)(ws_size / sizeof(float)) - 2LL * (long long)MAT;
  int P = (availElems > 0) ? (int)(availElems / (long long)MAT) : 1;
  if (P < 1) P = 1;
  if (P > BATCH) P = BATCH;

  build_A_kernel  <<<DIM, 256, 0, stream>>>(logd, lt, dA);
  build_cov_kernel<<<DIM, 256, 0, stream>>>(dA, dcov);

  for (int base = 0; base < BATCH; base += P) {
    int n = BATCH - base;
    if (n > P) n = P;
    chol_nll_kernel<<<n, 256, 0, stream>>>(dcov, x, mu, mask, out, slabs, base);
  }
}